// Attn_23605140259133
// MI455X (gfx1250) — compile-verified
//
#include <hip/hip_runtime.h>

#define EMBED 512
#define MAXLEN 2048
#define NBATCH 256

typedef __attribute__((ext_vector_type(2))) float v2f;
typedef __attribute__((ext_vector_type(8))) float v8f;

// ---------- wave32 reductions ----------
__device__ __forceinline__ float waveMax32(float x) {
#pragma unroll
  for (int off = 16; off > 0; off >>= 1)
    x = fmaxf(x, __shfl_xor(x, off, 32));
  return x;
}
__device__ __forceinline__ float waveSum32(float x) {
#pragma unroll
  for (int off = 16; off > 0; off >>= 1)
    x += __shfl_xor(x, off, 32);
  return x;
}

// ---------- Kernel 1: fold linear layer ----------
// v[d] = sum_e wv[e] * lin_w[e*EMBED + d]   (Wᵀ·wv)
// v[EMBED] = sum_e wv[e] * lin_b[e]          (scalar bias term)
__global__ __launch_bounds__(EMBED) void fold_kernel(
    const float* __restrict__ lin_w, const float* __restrict__ lin_b,
    const float* __restrict__ wv, float* __restrict__ vout) {
  __shared__ float sRed[16];
  const int d = threadIdx.x;
  float acc = 0.f;
#pragma unroll 4
  for (int e = 0; e < EMBED; ++e)
    acc = __builtin_fmaf(wv[e], lin_w[e * EMBED + d], acc);
  vout[d] = acc;

  float pc = wv[d] * lin_b[d];
  pc = waveSum32(pc);
  const int lane = d & 31, wid = d >> 5;
  if (lane == 0) sRed[wid] = pc;
  __syncthreads();
  if (wid == 0) {
    float y = (lane < 16) ? sRed[lane] : 0.f;
    y = waveSum32(y);
    if (lane == 0) vout[EMBED] = y;
  }
}

// ---------- Kernel 2: energies via WMMA + masked softmax ----------
// One block per batch row b. 512 threads = 16 waves.
// Each wave computes 16-token tiles: energy tile D = A(16x4 f32) x B(4x16 f32) + C
// with B's 16 columns all equal to v[k:k+4], so D's columns all hold the energies.
__global__ __launch_bounds__(512) void attn_energy_softmax(
    const float* __restrict__ q, const int* __restrict__ lens,
    const float* __restrict__ vfold, float* __restrict__ out) {
  __shared__ __align__(16) float sV[EMBED];
  __shared__ __align__(16) float sE[MAXLEN];
  __shared__ float sRed[16];
  __shared__ float sB[2];

  const int tid = threadIdx.x;
  const int b = blockIdx.x;
  const int lane = tid & 31;
  const int wid = tid >> 5;

  sV[tid] = vfold[tid];
  if (tid == 0) sB[0] = vfold[EMBED];
  const int len = lens[b];
  __syncthreads();
  const float cbias = sB[0];

  // fp32 WMMA A-operand layout: lane m = lane&15 is the row; lanes<16 carry
  // K={0,1}, lanes>=16 carry K={2,3}  ->  one b64 load per lane per K-step-4.
  const int mrow = lane & 15;
  const int koff = (lane >> 4) << 1;  // 0 or 2
  const float* qrow = q + (size_t)b * MAXLEN * EMBED;

  for (int t = wid; t < MAXLEN / 16; t += 16) {
    const float* aBase = qrow + (size_t)(t * 16 + mrow) * EMBED + koff;
    v8f acc0 = {};
    v8f acc1 = {};
    for (int k = 0; k < EMBED; k += 32) {
      v2f a[8], bb[8];
#pragma unroll
      for (int j = 0; j < 8; ++j)
        a[j] = *(const v2f*)(aBase + k + 4 * j);        // global_load_b64
#pragma unroll
      for (int j = 0; j < 8; ++j)
        bb[j] = *(const v2f*)(&sV[k + koff + 4 * j]);   // ds_load_b64
#pragma unroll
      for (int j = 0; j < 8; j += 2) {
        acc0 = __builtin_amdgcn_wmma_f32_16x16x4_f32(
            false, a[j],     false, bb[j],     (short)0, acc0, false, false);
        acc1 = __builtin_amdgcn_wmma_f32_16x16x4_f32(
            false, a[j + 1], false, bb[j + 1], (short)0, acc1, false, false);
      }
    }
    v8f acc = acc0 + acc1;
    // C/D layout: VGPR i holds M=i (lanes 0-15) / M=8+i (lanes 16-31).
    // All N columns identical -> lanes 0 and 16 publish the 16 energies.
    if ((lane & 15) == 0) {
      const int base = t * 16 + ((lane >> 4) << 3);
#pragma unroll
      for (int i = 0; i < 8; ++i) sE[base + i] = acc[i] + cbias;
    }
  }
  __syncthreads();

  // ----- masked softmax over sE[0..MAXLEN) -----
  const float NEG_INF = -__builtin_inff();
  const int l0 = tid * 4;
  float e[4];
  float lmax = NEG_INF;
#pragma unroll
  for (int i = 0; i < 4; ++i) {
    e[i] = (l0 + i < len) ? sE[l0 + i] : NEG_INF;
    lmax = fmaxf(lmax, e[i]);
  }
  lmax = waveMax32(lmax);
  if (lane == 0) sRed[wid] = lmax;
  __syncthreads();
  if (wid == 0) {
    float y = (lane < 16) ? sRed[lane] : NEG_INF;
    y = waveMax32(y);
    if (lane == 0) sB[1] = y;
  }
  __syncthreads();
  const float gmax = sB[1];

  float lsum = 0.f;
#pragma unroll
  for (int i = 0; i < 4; ++i) {
    const float ex = (l0 + i < len) ? __expf(e[i] - gmax) : 0.f;
    e[i] = ex;
    lsum += ex;
  }
  lsum = waveSum32(lsum);
  __syncthreads();  // all reads of sRed from max phase done
  if (lane == 0) sRed[wid] = lsum;
  __syncthreads();
  if (wid == 0) {
    float y = (lane < 16) ? sRed[lane] : 0.f;
    y = waveSum32(y);
    if (lane == 0) sB[0] = 1.0f / y;
  }
  __syncthreads();
  const float inv = sB[0];

  float4 r = make_float4(e[0] * inv, e[1] * inv, e[2] * inv, e[3] * inv);
  *(float4*)(out + (size_t)b * MAXLEN + l0) = r;
}

// ---------- host-side launch ----------
extern "C" void kernel_launch(void* const* d_in, const int* in_sizes, int n_in,
                              void* d_out, int out_size, void* d_ws, size_t ws_size,
                              hipStream_t stream) {
  const float* q     = (const float*)d_in[0];  // [256, 2048, 512] f32
  const int*   lens  = (const int*)d_in[1];    // [256] i32
  const float* lin_w = (const float*)d_in[2];  // [512, 512] f32
  const float* lin_b = (const float*)d_in[3];  // [512] f32
  const float* wv    = (const float*)d_in[4];  // [512] f32
  float* out = (float*)d_out;                  // [256, 2048] f32
  float* ws  = (float*)d_ws;                   // uses (EMBED+1) floats

  fold_kernel<<<1, EMBED, 0, stream>>>(lin_w, lin_b, wv, ws);
  attn_energy_softmax<<<NBATCH, 512, 0, stream>>>(q, lens, ws, out);
}